// GraphSAGEModel_68058051772434
// MI455X (gfx1250) — compile-verified
//
#include <hip/hip_runtime.h>

#define N_NODES 50000
#define N_EDGES 800000
#define F_IN    128
#define H1      256
#define H2      256
#define C_OUT   64
#define KC      32          // K-chunk staged into LDS per iteration

typedef float v2f __attribute__((ext_vector_type(2)));
typedef float v8f __attribute__((ext_vector_type(8)));

// D = A(16x4, fp32) * B(4x16, fp32) + C(16x16, fp32) on the matrix cores.
__device__ __forceinline__ v8f wmma_f32(v2f a, v2f b, v8f c) {
    return __builtin_amdgcn_wmma_f32_16x16x4_f32(
        /*neg_a=*/false, a, /*neg_b=*/false, b,
        /*c_mod=*/(short)0, c, /*reuse_a=*/false, /*reuse_b=*/false);
}

// Async global->LDS copy of one dword (gfx1250 GLOBAL_LOAD_ASYNC_TO_LDS_B32,
// tracked with ASYNCcnt). ldsOff is the wave-relative LDS byte address.
__device__ __forceinline__ void async_copy_b32(unsigned ldsOff, const float* g) {
    asm volatile("global_load_async_to_lds_b32 %0, %1, off"
                 :: "v"(ldsOff), "v"(g) : "memory");
}
__device__ __forceinline__ void wait_asynccnt0() {
    asm volatile("s_wait_asynccnt 0x0" ::: "memory");
}

// ---------------- degree / inverse-degree ----------------
__global__ void degree_kernel(const long long* __restrict__ ei,
                              float* __restrict__ cnt, int E) {
    int e = blockIdx.x * blockDim.x + threadIdx.x;
    if (e < E) atomicAdd(&cnt[(int)ei[(size_t)E + e]], 1.0f);
}

__global__ void inv_kernel(float* __restrict__ cnt, int n) {
    int i = blockIdx.x * blockDim.x + threadIdx.x;
    if (i < n) cnt[i] = 1.0f / fmaxf(cnt[i], 1.0f);
}

// ------- edge scatter-add: gather x[src] (float4), atomicAdd into agg[dst] ----
__global__ void scatter_kernel(const long long* __restrict__ ei,
                               const float* __restrict__ X,
                               float* __restrict__ agg, int E, int F) {
    int e    = blockIdx.x * (blockDim.x >> 5) + (threadIdx.x >> 5);
    int lane = threadIdx.x & 31;
    if (e >= E) return;
    int s = (int)ei[e];
    int d = (int)ei[(size_t)E + e];
    const float4* xs = (const float4*)(X + (size_t)s * F);
    float* ag        = agg + (size_t)d * F;
    for (int f4 = lane; f4 * 4 < F; f4 += 32) {
        float4 v = xs[f4];
        float* p = ag + f4 * 4;
        atomicAdd(p + 0, v.x);
        atomicAdd(p + 1, v.y);
        atomicAdd(p + 2, v.z);
        atomicAdd(p + 3, v.w);
    }
}

// ---------------- fused (mean*W_l + x*W_r + b) [+ReLU] via fp32 WMMA ----------
// Each wave computes a 16(M) x NT*16(N) output tile. Weights are staged in
// K-chunks of KC rows into LDS via async global->LDS copies, pair-interleaved
// so each B fragment is one aligned ds_load_b64. All waves reach barriers
// (out-of-range tiles are clamped: duplicate work, identical stores).
// A-fragment (32b, 16x4): lanes 0-15 -> K=k,k+1 ; lanes 16-31 -> K=k+2,k+3.
// C/D layout: col = lane&15 (+16*c), row = vgpr + 8*(lane>>4).
template <bool HAS_AGG, bool RELU, int NT>
__global__ void sage_gemm_kernel(const float* __restrict__ agg,
                                 const float* __restrict__ inv,
                                 const float* __restrict__ xr,
                                 const float* __restrict__ Wl,
                                 const float* __restrict__ Wr,
                                 const float* __restrict__ bias,
                                 float* __restrict__ out,
                                 int n, int K, int Nout) {
    constexpr int NC = NT * 16;                       // cols handled per block
    __shared__ float ldsbuf[KC * NC * (HAS_AGG ? 2 : 1)];
    float* ldsR = ldsbuf;                             // Wr chunk (pair-interleaved)
    float* ldsL = HAS_AGG ? (ldsbuf + KC * NC) : nullptr;

    int wave = threadIdx.x >> 5;
    int lane = threadIdx.x & 31;
    int nT   = n >> 4;
    int tile = blockIdx.x * (int)(blockDim.x >> 5) + wave;
    if (tile >= nT) tile = nT - 1;   // clamp: duplicate work, identical stores
    int row0 = tile << 4;
    int col0 = blockIdx.y * NC;
    int r    = lane & 15;
    int half = lane >> 4;

    const float* xrow = xr + (size_t)(row0 + r) * K + 2 * half;
    const float* arow = HAS_AGG ? (agg + (size_t)(row0 + r) * K + 2 * half) : nullptr;
    float scale       = HAS_AGG ? inv[row0 + r] : 0.0f;

    v8f acc[NT];
#pragma unroll
    for (int c = 0; c < NT; ++c) acc[c] = (v8f){};

    for (int k0 = 0; k0 < K; k0 += KC) {
        // ---- stage weight chunk rows [k0, k0+KC) into LDS, K-pair interleaved:
        // lds[(p*NC + col)*2 + j] = W[k0 + 2p + j][col0 + col]
        __syncthreads();                      // prior chunk fully consumed
        for (int idx = threadIdx.x; idx < (KC / 2) * NC; idx += blockDim.x) {
            int p  = idx / NC;
            int cc = idx - p * NC;
            const float* g = Wr + (size_t)(k0 + 2 * p) * Nout + col0 + cc;
            unsigned l0 = (unsigned)(uintptr_t)&ldsR[idx * 2];
            async_copy_b32(l0,      g);
            async_copy_b32(l0 + 4u, g + Nout);
            if (HAS_AGG) {
                const float* gl = Wl + (size_t)(k0 + 2 * p) * Nout + col0 + cc;
                unsigned l1 = (unsigned)(uintptr_t)&ldsL[idx * 2];
                async_copy_b32(l1,      gl);
                async_copy_b32(l1 + 4u, gl + Nout);
            }
        }
        wait_asynccnt0();                     // our async writes landed in LDS
        __syncthreads();                      // everyone's writes visible

        // ---- WMMA over the chunk: VMEM = 2x global_load_b64 (A) per k-step;
        // all NT B fragments are fetched first so the ds loads clause up behind
        // one wait and the WMMAs issue back-to-back.
        for (int k = 0; k < KC; k += 4) {
            v2f ax;
            ax.x = xrow[k0 + k];
            ax.y = xrow[k0 + k + 1];
            int bbase = ((k >> 1) + half) * NC * 2 + r * 2;

            v2f bR[NT];
#pragma unroll
            for (int c = 0; c < NT; ++c)
                bR[c] = *(const v2f*)&ldsR[bbase + c * 32];
#pragma unroll
            for (int c = 0; c < NT; ++c)
                acc[c] = wmma_f32(ax, bR[c], acc[c]);

            if (HAS_AGG) {
                v2f am;
                am.x = arow[k0 + k] * scale;
                am.y = arow[k0 + k + 1] * scale;
                v2f bL[NT];
#pragma unroll
                for (int c = 0; c < NT; ++c)
                    bL[c] = *(const v2f*)&ldsL[bbase + c * 32];
#pragma unroll
                for (int c = 0; c < NT; ++c)
                    acc[c] = wmma_f32(am, bL[c], acc[c]);
            }
        }
    }

    // ---- epilogue: bias (+ReLU), store 16 x NC tile
#pragma unroll
    for (int c = 0; c < NT; ++c) {
        int col  = col0 + c * 16 + r;
        float bi = bias[col];
#pragma unroll
        for (int v = 0; v < 8; v++) {
            int row = row0 + v + 8 * half;
            float o = acc[c][v] + bi;
            if (RELU) o = fmaxf(o, 0.0f);
            out[(size_t)row * Nout + col] = o;
        }
    }
}

extern "C" void kernel_launch(void* const* d_in, const int* in_sizes, int n_in,
                              void* d_out, int out_size, void* d_ws, size_t ws_size,
                              hipStream_t stream) {
    const float*     x   = (const float*)d_in[0];
    const long long* ei  = (const long long*)d_in[1];
    const float*     W1l = (const float*)d_in[2];
    const float*     b1  = (const float*)d_in[3];
    const float*     W1r = (const float*)d_in[4];
    const float*     W2l = (const float*)d_in[5];
    const float*     b2  = (const float*)d_in[6];
    const float*     W2r = (const float*)d_in[7];
    const float*     Wc  = (const float*)d_in[8];
    const float*     bc  = (const float*)d_in[9];
    float*           out = (float*)d_out;

    // Workspace layout (~180 MB)
    char*  ws  = (char*)d_ws;
    size_t off = 0;
    auto alloc = [&](size_t bytes) -> void* {
        void* p = ws + off;
        off += (bytes + 255) & ~(size_t)255;
        return p;
    };
    float* inv  = (float*)alloc((size_t)N_NODES * sizeof(float));
    float* agg1 = (float*)alloc((size_t)N_NODES * F_IN * sizeof(float));
    float* h1v  = (float*)alloc((size_t)N_NODES * H1 * sizeof(float));
    float* agg2 = (float*)alloc((size_t)N_NODES * H1 * sizeof(float));
    float* h2v  = (float*)alloc((size_t)N_NODES * H2 * sizeof(float));

    hipMemsetAsync(inv,  0, (size_t)N_NODES * sizeof(float), stream);
    hipMemsetAsync(agg1, 0, (size_t)N_NODES * F_IN * sizeof(float), stream);
    hipMemsetAsync(agg2, 0, (size_t)N_NODES * H1 * sizeof(float), stream);

    degree_kernel<<<(N_EDGES + 255) / 256, 256, 0, stream>>>(ei, inv, N_EDGES);
    inv_kernel<<<(N_NODES + 255) / 256, 256, 0, stream>>>(inv, N_NODES);

    const int nTiles  = N_NODES / 16;                // 3125 (N divisible by 16)
    const int mBlocks = (nTiles + 7) / 8;            // 8 waves per block

    // Layer 1: h1 = relu(mean1 @ W1_l + b1 + x @ W1_r)
    scatter_kernel<<<(N_EDGES + 7) / 8, 256, 0, stream>>>(ei, x, agg1, N_EDGES, F_IN);
    dim3 g1(mBlocks, H1 / 128);
    sage_gemm_kernel<true, true, 8><<<g1, 256, 0, stream>>>(
        agg1, inv, x, W1l, W1r, b1, h1v, N_NODES, F_IN, H1);

    // Layer 2: h2 = relu(mean2 @ W2_l + b2 + h1 @ W2_r)
    scatter_kernel<<<(N_EDGES + 7) / 8, 256, 0, stream>>>(ei, h1v, agg2, N_EDGES, H1);
    dim3 g2(mBlocks, H2 / 128);
    sage_gemm_kernel<true, true, 8><<<g2, 256, 0, stream>>>(
        agg2, inv, h1v, W2l, W2r, b2, h2v, N_NODES, H1, H2);

    // Classifier head: out = h2 @ Wc + bc (no aggregation path, no ReLU)
    dim3 g3(mBlocks, 1);
    sage_gemm_kernel<false, false, 4><<<g3, 256, 0, stream>>>(
        nullptr, nullptr, h2v, nullptr, Wc, bc, out, N_NODES, H2, C_OUT);
}